// NavierStokesSplittingEuler_77257871720615
// MI455X (gfx1250) — compile-verified
//
#include <hip/hip_runtime.h>
#include <stdint.h>

#if __has_builtin(__builtin_amdgcn_wmma_f32_16x16x4_f32)
#define HAS_WMMA_F32 1
#else
#define HAS_WMMA_F32 0
#endif
#if __has_builtin(__builtin_amdgcn_global_load_async_to_lds_b32)
#define HAS_ASYNC_LDS 1
#else
#define HAS_ASYNC_LDS 0
#endif

typedef float v2f __attribute__((ext_vector_type(2)));
typedef float v8f __attribute__((ext_vector_type(8)));

typedef __attribute__((address_space(1))) int* gptr_i32;
typedef __attribute__((address_space(3))) int* lptr_i32;

static constexpr int   NXc   = 512, NYc = 512, MASKc = 511;
static constexpr int   ROWS  = 32;             // owned x-rows per slab (512/16)
static constexpr int   LROWS = 34;             // + 2 halo rows
static constexpr int   SLABF = LROWS * NYc;    // floats per LDS buffer (17408)
static constexpr int   NWG   = 32;             // 2 channels * 16 slabs
static constexpr int   NTHREADS = 512;         // 16 wave32s
static constexpr int   NITER = 1000;

static constexpr float DTc = 0.1f, NUc = 0.1f, BETAc = 0.5f, RHOc = 1.0f,
                       FORCINGc = 0.0f, DXc = 1.0f;

// workspace layout (in floats)
static constexpr size_t WS_VSTAR   = 0;                       // [2][2][512][512]
static constexpr size_t WS_HALO    = (size_t)2*2*NXc*NYc;     // 1048576
static constexpr size_t WS_HALO_SZ = (size_t)2*2*16*2*NYc;    // parity*ch*slab*edge*y
static constexpr size_t WS_BAR     = WS_HALO + WS_HALO_SZ;    // 2 ints follow

__device__ __forceinline__ int wrapy(int y) { return y & MASKc; }

__device__ __forceinline__ void grid_barrier(int* cnt, int* gen) {
  __threadfence();
  __syncthreads();
  if (threadIdx.x == 0) {
    int g = __hip_atomic_load(gen, __ATOMIC_RELAXED, __HIP_MEMORY_SCOPE_AGENT);
    int v = __hip_atomic_fetch_add(cnt, 1, __ATOMIC_ACQ_REL, __HIP_MEMORY_SCOPE_AGENT);
    if (v == NWG - 1) {
      __hip_atomic_store(cnt, 0, __ATOMIC_RELAXED, __HIP_MEMORY_SCOPE_AGENT);
      __hip_atomic_fetch_add(gen, 1, __ATOMIC_RELEASE, __HIP_MEMORY_SCOPE_AGENT);
    } else {
      while (__hip_atomic_load(gen, __ATOMIC_ACQUIRE, __HIP_MEMORY_SCOPE_AGENT) == g) {
        __builtin_amdgcn_s_sleep(2);
      }
    }
  }
  __syncthreads();
  __threadfence();
}

__global__ void __launch_bounds__(NTHREADS, 1)
ns_fused(const float* __restrict__ V, const float* __restrict__ P,
         float* __restrict__ out, float* __restrict__ ws, int* __restrict__ bar)
{
  extern __shared__ float sm[];
  const int tid  = threadIdx.x;
  const int bid  = blockIdx.x;
  const int ch   = bid >> 4;
  const int slab = bid & 15;
  const int g0   = slab * ROWS;
  const int lane = tid & 31;
  const int wid  = tid >> 5;
  const int n    = lane & 15;
  const int hi   = lane >> 4;

  float* buf0  = sm;
  float* buf1  = sm + SLABF;
  float* vstar = ws + WS_VSTAR;
  float* halo  = ws + WS_HALO;

  const float* V0 = V + (size_t)(0*2 + ch) * NXc * NYc;
  const float* V1 = V + (size_t)(1*2 + ch) * NXc * NYc;
  const float* Pc = P + (size_t)ch * NXc * NYc;

  // ---------------- Prologue: V_star rows g0-1..g0+32 into LDS ----------------
  for (int r = 0; r < LROWS; ++r) {
    int gx = (g0 + r + (NXc - 1)) & MASKc;       // g0 + r - 1, periodic
    int xp = (gx + 1) & MASKc, xm = (gx + (NXc - 1)) & MASKc;
    int y = tid, yp = (y + 1) & MASKc, ym = (y + (NYc - 1)) & MASKc;
    float v0c  = V0[gx*NYc + y],  v1c  = V1[gx*NYc + y];
    float v0xp = V0[xp*NYc + y],  v0xm = V0[xm*NYc + y];
    float v0yp = V0[gx*NYc + yp], v0ym = V0[gx*NYc + ym];
    float v1xp = V1[xp*NYc + y],  v1xm = V1[xm*NYc + y];
    float v1yp = V1[gx*NYc + yp], v1ym = V1[gx*NYc + ym];
    float pxp  = Pc[xp*NYc + y],  pxm  = Pc[xm*NYc + y];
    float pyp  = Pc[gx*NYc + yp], pym  = Pc[gx*NYc + ym];
    float ddxV0 = 0.5f*(v0xp - v0xm), ddyV0 = 0.5f*(v0yp - v0ym);
    float ddxV1 = 0.5f*(v1xp - v1xm), ddyV1 = 0.5f*(v1yp - v1ym);
    float lap0  = v0xp + v0xm + v0yp + v0ym - 4.0f*v0c;
    float lap1  = v1xp + v1xm + v1yp + v1ym - 4.0f*v1c;
    float conv0 = v0c*ddxV0 + v1c*ddyV0;
    float conv1 = v0c*ddxV1 + v1c*ddyV1;
    float gpx = 0.5f*(pxp - pxm), gpy = 0.5f*(pyp - pym);
    buf0[r*NYc + y] = v0c + DTc*(-conv0 - (BETAc/RHOc)*gpx + NUc*lap0 + FORCINGc);
    buf1[r*NYc + y] = v1c + DTc*(-conv1 - (BETAc/RHOc)*gpy + NUc*lap1 + FORCINGc);
  }
  __syncthreads();

  // owned V_star rows -> workspace (needed by epilogue)
  for (int r = 1; r <= ROWS; ++r) {
    int gx = g0 + r - 1;
    vstar[(size_t)(0*2 + ch)*NXc*NYc + gx*NYc + tid] = buf0[r*NYc + tid];
    vstar[(size_t)(1*2 + ch)*NXc*NYc + gx*NYc + tid] = buf1[r*NYc + tid];
  }

  // ---- B*DX^2 in registers, in WMMA C/D fragment layout (4 tiles / wave) ----
  float bs[4][8];
  for (int k = 0; k < 4; ++k) {
    int tt = wid + 16*k;
    int tb = tt >> 5, ty = tt & 31;
    int x0 = 1 + 16*tb, y0 = 16*ty;
#pragma unroll
    for (int r = 0; r < 8; ++r) {
      int mm = r + 8*hi;
      int x = x0 + mm, yy = y0 + n;
      float dvx = 0.5f*(buf0[(x+1)*NYc + yy] - buf0[(x-1)*NYc + yy]);
      float dvy = 0.5f*(buf1[x*NYc + wrapy(yy+1)] - buf1[x*NYc + wrapy(yy-1)]);
      bs[k][r] = (RHOc/DTc) * (dvx + dvy) * (DXc*DXc);
    }
  }
  __syncthreads();

  // ---------------- phi(0) = P (owned rows + both halos) ----------------
  for (int r = 0; r < LROWS; ++r) {
    int gx = (g0 + r + (NXc - 1)) & MASKc;
    buf0[r*NYc + tid] = Pc[gx*NYc + tid];
  }
  __syncthreads();

#if HAS_WMMA_F32
  // Constant shift matrix S (1 on sub/super-diagonal) in B-fragment layout:
  // chunk c, VGPR j, lane L: K = 4c + j + 2*(L>>4), N = L&15
  v2f bfrag[4];
#pragma unroll
  for (int c4 = 0; c4 < 4; ++c4) {
    int k0 = 4*c4 + 2*hi;
    int k1 = k0 + 1;
    bfrag[c4].x = (n == k0 - 1 || n == k0 + 1) ? 1.0f : 0.0f;
    bfrag[c4].y = (n == k1 - 1 || n == k1 + 1) ? 1.0f : 0.0f;
  }
#endif

  // ---------------- 1000 Jacobi sweeps, phi resident in LDS ----------------
  int cur = 0;
  for (int it = 0; it < NITER; ++it) {
    const float* bc = sm + cur * SLABF;
    float*       bn = sm + (1 - cur) * SLABF;

    for (int k = 0; k < 4; ++k) {
      int tt = wid + 16*k;
      int tb = tt >> 5, ty = tt & 31;
      int x0 = 1 + 16*tb, y0 = 16*ty;
      int yA = y0 + n;
      int yl = (y0 + (NYc - 1)) & MASKc;   // y0-1 (periodic)
      int yr = (y0 + 16) & MASKc;          // y0+16 (periodic)
#if HAS_WMMA_F32
      v8f acc;
#pragma unroll
      for (int r = 0; r < 8; ++r) {
        int mm = r + 8*hi, x = x0 + mm;
        float up = bc[(x-1)*NYc + yA];
        float dn = bc[(x+1)*NYc + yA];
        int   eo = (n == 0) ? yl : yr;
        float ev = bc[x*NYc + eo];
        ev = (n == 0 || n == 15) ? ev : 0.0f;  // cross-tile y-neighbor fixup
        acc[r] = up + dn + ev - bs[k][r];
      }
      // D = A(phi tile) x S + C  via four K=4 chunks
#pragma unroll
      for (int c4 = 0; c4 < 4; ++c4) {
        int kk  = 4*c4 + 2*hi;
        int row = x0 + n;                   // A-fragment: M = lane&15
        v2f a;
        a.x = bc[row*NYc + y0 + kk];
        a.y = bc[row*NYc + y0 + kk + 1];
        acc = __builtin_amdgcn_wmma_f32_16x16x4_f32(
                  false, a, false, bfrag[c4], (short)0, acc, false, false);
      }
#pragma unroll
      for (int r = 0; r < 8; ++r) {
        int mm = r + 8*hi, x = x0 + mm;
        bn[x*NYc + yA] = 0.25f * acc[r];
      }
#else
#pragma unroll
      for (int r = 0; r < 8; ++r) {
        int mm = r + 8*hi, x = x0 + mm;
        float up = bc[(x-1)*NYc + yA], dn = bc[(x+1)*NYc + yA];
        float lf = bc[x*NYc + wrapy(yA-1)], rt = bc[x*NYc + wrapy(yA+1)];
        bn[x*NYc + yA] = 0.25f*(up + dn + lf + rt - bs[k][r]);
      }
#endif
    }
    __syncthreads();

    // publish new edge rows (parity double-buffered halo exchange)
    int p = it & 1;
    float* hp = halo + ((size_t)p * 2 + ch) * (16 * 2 * NYc);
    hp[(slab*2 + 0)*NYc + tid] = bn[1*NYc + tid];
    hp[(slab*2 + 1)*NYc + tid] = bn[ROWS*NYc + tid];

    grid_barrier(bar, bar + 1);

    // fetch neighbor halos into rows 0 / 33 of the new buffer
    int sUp = (slab + 15) & 15, sDn = (slab + 1) & 15;
    float* src0 = const_cast<float*>(&hp[(sUp*2 + 1)*NYc + tid]);
    float* src1 = const_cast<float*>(&hp[(sDn*2 + 0)*NYc + tid]);
#if HAS_ASYNC_LDS
    {
      gptr_i32 gp0 = (gptr_i32)(void*)src0;
      lptr_i32 lp0 = (lptr_i32)(void*)&bn[0*NYc + tid];
      __builtin_amdgcn_global_load_async_to_lds_b32(gp0, lp0, 0, 0);
      gptr_i32 gp1 = (gptr_i32)(void*)src1;
      lptr_i32 lp1 = (lptr_i32)(void*)&bn[(LROWS-1)*NYc + tid];
      __builtin_amdgcn_global_load_async_to_lds_b32(gp1, lp1, 0, 0);
#if __has_builtin(__builtin_amdgcn_s_wait_asynccnt)
      __builtin_amdgcn_s_wait_asynccnt(0);
#else
      asm volatile("s_wait_asynccnt 0" ::: "memory");
#endif
    }
#else
    bn[0*NYc + tid]           = *src0;
    bn[(LROWS-1)*NYc + tid]   = *src1;
#endif
    __syncthreads();
    cur = 1 - cur;
  }

  // ---------------- Epilogue: V_new, P_new ----------------
  const float* phi = sm + cur * SLABF;
  for (int r = 1; r <= ROWS; ++r) {
    int gx = g0 + r - 1;
    float ph  = phi[r*NYc + tid];
    float gxv = 0.5f*(phi[(r+1)*NYc + tid] - phi[(r-1)*NYc + tid]);
    float gyv = 0.5f*(phi[r*NYc + wrapy(tid+1)] - phi[r*NYc + wrapy(tid-1)]);
    float vs0 = vstar[(size_t)(0*2 + ch)*NXc*NYc + gx*NYc + tid];
    float vs1 = vstar[(size_t)(1*2 + ch)*NXc*NYc + gx*NYc + tid];
    out[(size_t)(0*2 + ch)*NXc*NYc + gx*NYc + tid] = vs0 - (DTc/RHOc)*gxv;
    out[(size_t)(1*2 + ch)*NXc*NYc + gx*NYc + tid] = vs1 - (DTc/RHOc)*gyv;
    out[(size_t)4*NXc*NYc + (size_t)ch*NXc*NYc + gx*NYc + tid] = ph + BETAc*Pc[gx*NYc + tid];
  }
}

extern "C" void kernel_launch(void* const* d_in, const int* in_sizes, int n_in,
                              void* d_out, int out_size, void* d_ws, size_t ws_size,
                              hipStream_t stream) {
  (void)in_sizes; (void)n_in; (void)out_size; (void)ws_size;
  const float* V  = (const float*)d_in[0];   // [2,2,512,512]
  const float* P  = (const float*)d_in[1];   // [2,512,512]
  float* out = (float*)d_out;
  float* ws  = (float*)d_ws;
  int*   bar = (int*)(ws + WS_BAR);

  // reset device-barrier state every call (captured in graph -> deterministic)
  (void)hipMemsetAsync(bar, 0, 2 * sizeof(int), stream);

  size_t shmem = (size_t)2 * SLABF * sizeof(float);  // 139264 B LDS
  (void)hipFuncSetAttribute(reinterpret_cast<const void*>(ns_fused),
                            hipFuncAttributeMaxDynamicSharedMemorySize, (int)shmem);
  hipLaunchKernelGGL(ns_fused, dim3(NWG), dim3(NTHREADS), shmem, stream,
                     V, P, out, ws, bar);
}